// SSIM_41223096107661
// MI455X (gfx1250) — compile-verified
//
#include <hip/hip_runtime.h>

typedef float v2f __attribute__((ext_vector_type(2)));
typedef float v8f __attribute__((ext_vector_type(8)));

#define IMG_H 512
#define IMG_W 512
#define KCH 7              // 7 chunks of K=4 -> K=28 (covers band K=26)
#define RROWS 26           // halo rows per 16x16 tile
#define RSTRIDE 29         // 28 used cols + 1 pad (bank-conflict avoidance)
#define TSTRIDE 17         // 16 cols + 1 pad
#define TROWS 28           // rows 0..25 valid, 26..27 zeroed
#define IMG_FLOATS (RROWS * RSTRIDE)                    // 754
#define WAVE_FLOATS (2 * IMG_FLOATS + TROWS * TSTRIDE)  // 1984

// 11-tap Gaussian, sigma=1.5, normalized (matches reference _make_window)
__device__ const float GAUSS[11] = {
    0.0010284144f, 0.0075987582f, 0.0360008036f, 0.1093608653f, 0.2130046849f,
    0.2660147470f, 0.2130046849f, 0.1093608653f, 0.0360008036f, 0.0075987582f,
    0.0010284144f};

// banded weight: W[j][i] = g[j-i] when 0 <= j-i <= 10, else 0
__device__ __forceinline__ float wband(int j, int i) {
    unsigned d = (unsigned)(j - i);
    return (d < 11u) ? GAUSS[d] : 0.0f;
}

__global__ __launch_bounds__(256) void ssim_main_kernel(
    const float* __restrict__ img1, const float* __restrict__ img2,
    float* __restrict__ partial) {
    __shared__ float smem[8 * WAVE_FLOATS];
    __shared__ float redbuf[8];

    const int lane = threadIdx.x & 31;
    const int wave = threadIdx.x >> 5;
    const int tile = blockIdx.x * 8 + wave;  // 49152 tiles total

    // tile -> (plane, tile_y, tile_x); 32x32 tiles per 512x512 plane
    const int plane = tile >> 10;
    const int trem  = tile & 1023;
    const int ty = trem >> 5, tx = trem & 31;
    const int row0 = ty * 16, col0 = tx * 16;
    const size_t pbase = (size_t)plane * (size_t)(IMG_H * IMG_W);

    float* lds1 = smem + wave * WAVE_FLOATS;
    float* lds2 = lds1 + IMG_FLOATS;
    float* ldsT = lds2 + IMG_FLOATS;

    // ---- stage halo region (rows row0-5..row0+20, cols col0-5..col0+20) ----
    for (int idx = lane; idx < RROWS * 28; idx += 32) {
        const int r = idx / 28, c = idx % 28;
        const int gr = row0 - 5 + r;
        const int gc = col0 - 5 + c;
        float v1 = 0.0f, v2 = 0.0f;
        if (c < 26 && gr >= 0 && gr < IMG_H && gc >= 0 && gc < IMG_W) {
            const size_t goff = pbase + (size_t)gr * IMG_W + (size_t)gc;
            v1 = img1[goff];
            v2 = img2[goff];
        }
        lds1[r * RSTRIDE + c] = v1;
        lds2[r * RSTRIDE + c] = v2;
    }
    // zero T pad rows 26..27 (read by the K=28-padded vertical pass)
    {
        const int zr = 26 + (lane >> 4);
        const int zc = lane & 15;
        ldsT[zr * TSTRIDE + zc] = 0.0f;
    }

    const int half = lane >> 4;   // 0: K rows j0,j0+1 ; 1: K rows j0+2,j0+3
    const int l16  = lane & 15;

    // Banded Gaussian operand chunks. Lane mapping is identical for use as
    // the horizontal-pass B matrix and the vertical-pass A matrix.
    v2f wv2[KCH];
#pragma unroll
    for (int k = 0; k < KCH; ++k) {
        const int j0 = 4 * k + 2 * half;
        wv2[k].x = wband(j0, l16);
        wv2[k].y = wband(j0 + 1, l16);
    }

    // Separable 11x11 conv of one data plane via V_WMMA_F32_16X16X4_F32.
    auto conv_plane = [&](auto getval) -> v8f {
        // ---- horizontal pass: T[r][x] = sum_j In[r][j] * Wh[j][x] ----
        v8f t0 = {};  // region rows 0..15
        v8f t1 = {};  // region rows 10..25
#pragma unroll
        for (int k = 0; k < KCH; ++k) {
            const int c0c = 4 * k + 2 * half;
            v2f a0, a1;
            a0.x = getval(l16, c0c);
            a0.y = getval(l16, c0c + 1);
            a1.x = getval(10 + l16, c0c);
            a1.y = getval(10 + l16, c0c + 1);
            t0 = __builtin_amdgcn_wmma_f32_16x16x4_f32(false, a0, false, wv2[k],
                                                       (short)0, t0, false, false);
            t1 = __builtin_amdgcn_wmma_f32_16x16x4_f32(false, a1, false, wv2[k],
                                                       (short)0, t1, false, false);
        }
        // scatter D-layout -> LDS T (row-major), cross-lane relayout for B
#pragma unroll
        for (int v = 0; v < 8; ++v) {
            const int m0 = v + 8 * half;
            ldsT[m0 * TSTRIDE + l16] = t0[v];
            const int m1 = 10 + v + 8 * half;
            ldsT[m1 * TSTRIDE + l16] = t1[v];
        }
        // ---- vertical pass: Out[y][x] = sum_j Wv[y][j] * T[j][x] ----
        v8f out = {};
#pragma unroll
        for (int k = 0; k < KCH; ++k) {
            const int j0 = 4 * k + 2 * half;
            v2f b;
            b.x = ldsT[j0 * TSTRIDE + l16];
            b.y = ldsT[(j0 + 1) * TSTRIDE + l16];
            out = __builtin_amdgcn_wmma_f32_16x16x4_f32(false, wv2[k], false, b,
                                                        (short)0, out, false, false);
        }
        return out;
    };

    auto at1 = [&](int r, int c) { return lds1[r * RSTRIDE + c]; };
    auto at2 = [&](int r, int c) { return lds2[r * RSTRIDE + c]; };

    const v8f mu1 = conv_plane([&](int r, int c) { return at1(r, c); });
    const v8f mu2 = conv_plane([&](int r, int c) { return at2(r, c); });
    const v8f s11 = conv_plane([&](int r, int c) { float v = at1(r, c); return v * v; });
    const v8f s22 = conv_plane([&](int r, int c) { float v = at2(r, c); return v * v; });
    const v8f s12 = conv_plane([&](int r, int c) { return at1(r, c) * at2(r, c); });

    // ---- elementwise SSIM + local accumulation ----
    float acc = 0.0f;
    const float C1 = 1e-4f;   // 0.01^2
    const float C2 = 9e-4f;   // 0.03^2
#pragma unroll
    for (int v = 0; v < 8; ++v) {
        const float m1 = mu1[v], m2 = mu2[v];
        const float m1s = m1 * m1, m2s = m2 * m2, m12 = m1 * m2;
        const float sg1 = s11[v] - m1s;
        const float sg2 = s22[v] - m2s;
        const float sgx = s12[v] - m12;
        const float num = (2.0f * m12 + C1) * (2.0f * sgx + C2);
        const float den = (m1s + m2s + C1) * (sg1 + sg2 + C2);
        acc += num / den;
    }

    // wave32 shuffle reduction
#pragma unroll
    for (int off = 16; off > 0; off >>= 1)
        acc += __shfl_down(acc, off, 32);
    if (lane == 0) redbuf[wave] = acc;
    __syncthreads();
    if (threadIdx.x == 0) {
        float s = 0.0f;
#pragma unroll
        for (int w = 0; w < 8; ++w) s += redbuf[w];
        partial[blockIdx.x] = s;
    }
}

__global__ __launch_bounds__(256) void ssim_reduce_kernel(
    const float* __restrict__ partial, int n, float* __restrict__ out,
    float inv_n) {
    __shared__ float sm[256];
    float s = 0.0f;
    for (int i = threadIdx.x; i < n; i += 256) s += partial[i];
    sm[threadIdx.x] = s;
    __syncthreads();
    for (int off = 128; off > 0; off >>= 1) {
        if ((int)threadIdx.x < off) sm[threadIdx.x] += sm[threadIdx.x + off];
        __syncthreads();
    }
    if (threadIdx.x == 0) out[0] = sm[0] * inv_n;
}

extern "C" void kernel_launch(void* const* d_in, const int* in_sizes, int n_in,
                              void* d_out, int out_size, void* d_ws,
                              size_t ws_size, hipStream_t stream) {
    (void)in_sizes; (void)n_in; (void)out_size; (void)ws_size;
    const float* img1 = (const float*)d_in[0];
    const float* img2 = (const float*)d_in[1];
    float* partial = (float*)d_ws;

    const int nTiles = 16 * 3 * (IMG_H / 16) * (IMG_W / 16);  // 49152
    const int blocks = nTiles / 8;                            // 6144

    ssim_main_kernel<<<blocks, 256, 0, stream>>>(img1, img2, partial);

    const float inv_n = 1.0f / (16.0f * 3.0f * 512.0f * 512.0f);
    ssim_reduce_kernel<<<1, 256, 0, stream>>>(partial, blocks, (float*)d_out,
                                              inv_n);
}